// SampleAndGroup_70446053589570
// MI455X (gfx1250) — compile-verified
//
#include <hip/hip_runtime.h>

// ---------------- problem constants ----------------
#define NPTS    16384
#define NCENT   1024
#define NSAMP   64
#define NCH     64
#define NBATCH  8
#define R2      0.04f        // RADIUS^2 = 0.2^2
#define NTILE   (NPTS / 16)  // 1024 WMMA point-tiles per batch row

// ---------------- output layout (flat float offsets) ----------------
// new_xyz      : 8*1024*3              @ 0
// new_points   : 8*1024*64*67          @ 24576
// centroid_idx : 8*1024 (float-cast)   @ 35151872
// grouped_xyz  : 8*1024*64*3           @ 35160064
#define O_NEWXYZ   0L
#define O_NEWPTS   24576L
#define O_CENTIDX  35151872L
#define O_GRPXYZ   35160064L

typedef float v2f __attribute__((ext_vector_type(2)));
typedef float v8f __attribute__((ext_vector_type(8)));

// u64 max across a wave32 via ds_swizzle xor-butterfly (imm = xor<<10 | and=0x1f)
#define WMAX_STEP(K, IMM)                                                      \
  do {                                                                         \
    int _lo = __builtin_amdgcn_ds_swizzle((int)((K) & 0xffffffffULL), (IMM));  \
    int _hi = __builtin_amdgcn_ds_swizzle((int)((K) >> 32), (IMM));            \
    unsigned long long _o =                                                    \
        (((unsigned long long)(unsigned)_hi) << 32) | (unsigned)_lo;           \
    if (_o > (K)) (K) = _o;                                                    \
  } while (0)

#define WAVE_MAX64(K)   \
  WMAX_STEP(K, 0x041f); \
  WMAX_STEP(K, 0x081f); \
  WMAX_STEP(K, 0x101f); \
  WMAX_STEP(K, 0x201f); \
  WMAX_STEP(K, 0x401f)

// =====================================================================
// Kernel 1: farthest point sampling. One workgroup per batch.
// xyz staged as padded float4 AoS in 256KB dynamic LDS (WGP has 320KB)
// so each point fetch is one aligned wide ds_load. Each of 1024 threads
// owns 16 points; min_dist lives in VGPRs. Argmax key packs
// (f32 dist bits << 32) | (16383 - idx) so u64-max gives "max dist,
// first index on ties" == jnp.argmax semantics.
// =====================================================================
__global__ __launch_bounds__(1024) void fps_kernel(
    const float* __restrict__ xyz, int* __restrict__ centws,
    float* __restrict__ out) {
  extern __shared__ char smem[];
  float* Pf = (float*)smem;                                     // NPTS float4
  unsigned long long* scratch =
      (unsigned long long*)(smem + (size_t)NPTS * 16);          // [32]
  unsigned long long* bcast = scratch + 32;

  const int b = blockIdx.x;
  const int tid = threadIdx.x;

  // cooperative staging: xyz (p,c) -> Pf[p*4 + c]; pad word unused
  const float* src = xyz + (long)b * NPTS * 3;
  for (int i = tid; i < NPTS * 3; i += 1024) {
    float v = src[i];
    int p = i / 3;
    int c = i - p * 3;
    Pf[(p << 2) + c] = v;
  }
  __syncthreads();

  float md[16];
#pragma unroll
  for (int j = 0; j < 16; ++j) md[j] = 1e10f;  // matches reference init

  int cur = 0;
  for (int it = 0; it < NCENT; ++it) {
    if (tid == 0) {
      centws[b * NCENT + it] = cur;                   // for kernel 2
      out[O_CENTIDX + (long)b * NCENT + it] = (float)cur;
    }
    const float cx = Pf[(cur << 2) + 0];
    const float cy = Pf[(cur << 2) + 1];
    const float cz = Pf[(cur << 2) + 2];

    unsigned long long key = 0;
#pragma unroll
    for (int j = 0; j < 16; ++j) {
      int p = tid + (j << 10);
      float dx = Pf[(p << 2) + 0] - cx;
      float dy = Pf[(p << 2) + 1] - cy;
      float dz = Pf[(p << 2) + 2] - cz;
      float d = dx * dx + dy * dy + dz * dz;
      float m = md[j] < d ? md[j] : d;
      md[j] = m;
      unsigned long long k =
          ((unsigned long long)__float_as_uint(m) << 32) |
          (unsigned)(16383 - p);
      if (k > key) key = k;
    }

    WAVE_MAX64(key);                       // wave32 butterfly reduction
    if ((tid & 31) == 0) scratch[tid >> 5] = key;
    __syncthreads();
    if (tid < 32) {                        // 32 waves -> final reduce in wave 0
      unsigned long long k2 = scratch[tid];
      WAVE_MAX64(k2);
      if (tid == 0) *bcast = k2;
    }
    __syncthreads();
    cur = 16383 - (int)(unsigned)(*bcast & 0xffffffffULL);
  }
}

// =====================================================================
// Kernel 2: ball query via WMMA + grouping/gather.
// Grid: 8 batches x 64 centroid-tiles (16 centroids each), 512 thr (16 waves).
// Phase A (software-pipelined): each wave sweeps point tiles; the next
//   tile's coords are prefetched while V_WMMA_F32_16X16X4_F32 computes
//   D[s][n] = |p|^2 - 2 c.p for the current 16x16 tile. Per-row
//   threshold compare + ballot_w32 turns D rows into 32-bit masks
//   (lo16 = row r, hi16 = row r+8) stored tile-major; lane 0 commits
//   them as two ds_store_b128 (ballots are wave-uniform SGPRs).
// Phase B: 16 threads serially bit-scan masks -> first 64 indices (+pad).
// Phase C: wave-per-centroid gathers features (coalesced 256B rows);
//   lanes 0-2 write the relative-xyz components in parallel.
// =====================================================================
__global__ __launch_bounds__(512) void group_kernel(
    const float* __restrict__ xyz, const float* __restrict__ feat,
    const int* __restrict__ centws, float* __restrict__ out) {
  __shared__ unsigned smask[NTILE][8];  // 32KB: [tile][r], lo16=M=r hi16=M=r+8
  __shared__ int sidx[16][NSAMP];       // 4KB selected indices
  __shared__ float sC[4][16];           // cx,cy,cz, threshold

  const int b = blockIdx.x >> 6;
  const int tileBase = (blockIdx.x & 63) << 4;
  const int tid = threadIdx.x;
  const int lane = tid & 31;
  const int wid = tid >> 5;
  const float* xyzb = xyz + (long)b * NPTS * 3;

  if (tid < 16) {
    int ci = centws[b * NCENT + tileBase + tid];
    float cx = xyzb[3 * ci], cy = xyzb[3 * ci + 1], cz = xyzb[3 * ci + 2];
    sC[0][tid] = cx;
    sC[1][tid] = cy;
    sC[2][tid] = cz;
    sC[3][tid] = R2 - (cx * cx + cy * cy + cz * cz);  // d2<R2 <=> D < this
  }
  __syncthreads();

  const bool lo = lane < 16;
  const int h = lane & 15;

  // A-matrix (16x4 f32, 2 VGPRs): lanes 0-15 hold K=0,1; lanes 16-31 K=2,3
  v2f A;
  A.x = lo ? -2.0f * sC[0][h] : -2.0f * sC[2][h];
  A.y = lo ? -2.0f * sC[1][h] : 1.0f;

  // per-lane thresholds for D rows (hoisted out of the tile loop)
  float tsel[8];
#pragma unroll
  for (int r = 0; r < 8; ++r) tsel[r] = lo ? sC[3][r] : sC[3][r + 8];

  // Phase A: uniform per-wave tile loop (EXEC all-1s at the WMMA),
  // next-tile load overlapped with current-tile WMMA + ballot + store.
  {
    const float* p0 = xyzb + 3 * (wid * 16 + h);
    float px = p0[0], py = p0[1], pz = p0[2];
    for (int tile = wid; tile < NTILE; tile += 16) {
      int nt = tile + 16;
      const float* pn = xyzb + 3 * (((nt < NTILE) ? nt : tile) * 16 + h);
      float npx = pn[0], npy = pn[1], npz = pn[2];  // prefetch next tile

      // B-matrix (4x16 f32): lanes 0-15 hold K=0,1; lanes 16-31 hold K=2,3
      v2f Bv;
      Bv.x = lo ? px : pz;
      Bv.y = lo ? py : (px * px + py * py + pz * pz);
      v8f d = {};
      d = __builtin_amdgcn_wmma_f32_16x16x4_f32(false, A, false, Bv, (short)0,
                                                d, false, false);
      unsigned mb0 = __builtin_amdgcn_ballot_w32(d[0] < tsel[0]);
      unsigned mb1 = __builtin_amdgcn_ballot_w32(d[1] < tsel[1]);
      unsigned mb2 = __builtin_amdgcn_ballot_w32(d[2] < tsel[2]);
      unsigned mb3 = __builtin_amdgcn_ballot_w32(d[3] < tsel[3]);
      unsigned mb4 = __builtin_amdgcn_ballot_w32(d[4] < tsel[4]);
      unsigned mb5 = __builtin_amdgcn_ballot_w32(d[5] < tsel[5]);
      unsigned mb6 = __builtin_amdgcn_ballot_w32(d[6] < tsel[6]);
      unsigned mb7 = __builtin_amdgcn_ballot_w32(d[7] < tsel[7]);
      if (lane == 0) {  // wave-uniform SGPRs -> 2x ds_store_b128
        uint4* q = (uint4*)&smask[tile][0];
        q[0] = make_uint4(mb0, mb1, mb2, mb3);
        q[1] = make_uint4(mb4, mb5, mb6, mb7);
      }
      px = npx; py = npy; pz = npz;
    }
  }
  __syncthreads();

  // Phase B: first NSAMP in-radius indices in ascending order, pad w/ first
  if (tid < 16) {
    int cnt = 0;
    for (int w = 0; w < NTILE && cnt < NSAMP; ++w) {
      unsigned word = smask[w][tid & 7];
      unsigned m = (tid < 8) ? (word & 0xffffu) : (word >> 16);
      while (m && cnt < NSAMP) {
        int bpos = __ffs(m) - 1;
        sidx[tid][cnt++] = w * 16 + bpos;
        m &= m - 1;
      }
    }
    int first = (cnt > 0) ? sidx[tid][0] : 0;  // none found -> 0 (reference)
    for (int k = cnt; k < NSAMP; ++k) sidx[tid][k] = first;

    long o = O_NEWXYZ + (long)(b * NCENT + tileBase + tid) * 3;
    out[o + 0] = sC[0][tid];
    out[o + 1] = sC[1][tid];
    out[o + 2] = sC[2][tid];
  }
  __syncthreads();

  // Phase C: wave-per-centroid gather + write
  {
    const int s = tileBase + wid;
    const float cx = sC[0][wid], cy = sC[1][wid], cz = sC[2][wid];
    const long rowBase = (long)(b * NCENT + s) * NSAMP;
    for (int k = 0; k < NSAMP; ++k) {
      int j = sidx[wid][k];
      const float* frow = feat + ((long)b * NPTS + j) * NCH;
      long ro = O_NEWPTS + (rowBase + k) * 67;
      out[ro + 3 + lane] = frow[lane];         // channels 3..34
      out[ro + 35 + lane] = frow[32 + lane];   // channels 35..66
      if (lane < 3) {                          // one xyz component per lane
        float cc = (lane == 0) ? cx : ((lane == 1) ? cy : cz);
        float dv = xyzb[3 * j + lane] - cc;
        out[ro + lane] = dv;
        long go = O_GRPXYZ + (rowBase + k) * 3;
        out[go + lane] = dv;
      }
    }
  }
}

extern "C" void kernel_launch(void* const* d_in, const int* in_sizes, int n_in,
                              void* d_out, int out_size, void* d_ws,
                              size_t ws_size, hipStream_t stream) {
  (void)in_sizes; (void)n_in; (void)out_size; (void)ws_size;
  const float* xyz = (const float*)d_in[0];   // (8,16384,3) f32
  const float* feat = (const float*)d_in[1];  // (8,16384,64) f32
  float* out = (float*)d_out;
  int* centws = (int*)d_ws;  // 8*1024 int32 centroid indices

  const size_t smem =
      (size_t)NPTS * 16 + 33 * sizeof(unsigned long long);
  (void)hipFuncSetAttribute((const void*)fps_kernel,
                            hipFuncAttributeMaxDynamicSharedMemorySize,
                            (int)smem);

  fps_kernel<<<NBATCH, 1024, smem, stream>>>(xyz, centws, out);
  group_kernel<<<NBATCH * 64, 512, 0, stream>>>(xyz, feat, centws, out);
}